// GNN_52767968199327
// MI455X (gfx1250) — compile-verified
//
#include <hip/hip_runtime.h>

typedef __attribute__((ext_vector_type(2))) float v2f;
typedef __attribute__((ext_vector_type(8))) float v8f;

static __device__ __forceinline__ v8f wmma_f32_k4(v2f a, v2f b, v8f c) {
    // V_WMMA_F32_16X16X4_F32: D = A(16x4) x B(4x16) + C(16x16), fp32, wave32
    return __builtin_amdgcn_wmma_f32_16x16x4_f32(
        /*neg_a=*/false, a, /*neg_b=*/false, b,
        /*c_mod=*/(short)0, c, /*reuse_a=*/false, /*reuse_b=*/false);
}

// ---- norm precompute ------------------------------------------------------

__global__ void k_init(float* __restrict__ deg, float* __restrict__ pooled,
                       int NN, int PG) {
    int i = blockIdx.x * blockDim.x + threadIdx.x;
    if (i < NN) deg[i] = 1.0f;        // self-loop weight folded in
    if (i < PG) pooled[i] = 0.0f;     // zero pooled accumulator every call
}

__global__ void k_deg(const int* __restrict__ col, const float* __restrict__ w,
                      float* __restrict__ deg, int NE) {
    int e = blockIdx.x * blockDim.x + threadIdx.x;
    if (e < NE) atomicAdd(&deg[col[e]], w[e]);
}

__global__ void k_dinv(float* __restrict__ deg, int NN) {
    int i = blockIdx.x * blockDim.x + threadIdx.x;
    if (i < NN) {
        float d = deg[i];
        deg[i] = (d > 0.0f) ? rsqrtf(d) : 0.0f;   // in place: deg -> dinv
    }
}

__global__ void k_norm(const int* __restrict__ row, const int* __restrict__ col,
                       const float* __restrict__ w, const float* __restrict__ dinv,
                       float* __restrict__ nrm, int NE) {
    int e = blockIdx.x * blockDim.x + threadIdx.x;
    if (e < NE) nrm[e] = dinv[row[e]] * w[e] * dinv[col[e]];
}

// ---- dense layers via fp32 WMMA (complete 16-node tiles only) -------------

// hw[i,0:16] = x[i,0:3] @ W1[3,16]  (K padded 3 -> 4 with zeros), 1 WMMA/tile
__global__ void k_dense1(const float* __restrict__ x, const float* __restrict__ W1,
                         float* __restrict__ hw, int ntiles) {
    int wave = (blockIdx.x * blockDim.x + threadIdx.x) >> 5;
    if (wave >= ntiles) return;                       // wave-uniform guard
    int lane = threadIdx.x & 31;
    int half = lane >> 4;                             // K group: 0 -> K{0,1}, 1 -> K{2,3}
    int m    = lane & 15;
    int node = wave * 16 + m;
    // unconditional loads, value selects (no EXEC split before WMMA)
    float x0 = x[node * 3 + 0];
    float x1 = x[node * 3 + 1];
    float x2 = x[node * 3 + 2];
    float w0 = W1[0 * 16 + m];
    float w1 = W1[1 * 16 + m];
    float w2 = W1[2 * 16 + m];
    v2f a, b;
    a.x = half ? x2 : x0;
    a.y = half ? 0.0f : x1;                           // K=3 column padded to zero
    b.x = half ? w2 : w0;
    b.y = half ? 0.0f : w1;                           // B row 3 = 0
    v8f c = {};
    c = wmma_f32_k4(a, b, c);
    int base = (wave * 16 + half * 8) * 16 + m;
#pragma unroll
    for (int r = 0; r < 8; ++r)
        hw[base + r * 16] = c[r];                     // s_clause'd 64B row stores
}

// hw[i,0:16] = h1[i,0:16] @ W2[16,16], 4 chained WMMAs/tile
__global__ void k_dense2(const float* __restrict__ h1, const float* __restrict__ W2,
                         float* __restrict__ hw, int ntiles) {
    int wave = (blockIdx.x * blockDim.x + threadIdx.x) >> 5;
    if (wave >= ntiles) return;
    int lane = threadIdx.x & 31;
    int half = lane >> 4;
    int m    = lane & 15;
    int node = wave * 16 + m;
    const float2* __restrict__ h1v = (const float2*)h1;   // 8B-aligned pairs
    v8f c = {};
#pragma unroll
    for (int kb = 0; kb < 16; kb += 4) {
        int k = kb + half * 2;                        // even -> aligned float2
        float2 ap = h1v[(node * 16 + k) >> 1];        // global_load_b64
        v2f a, b;
        a.x = ap.x;
        a.y = ap.y;
        b.x = W2[k * 16 + m];
        b.y = W2[(k + 1) * 16 + m];
        c = wmma_f32_k4(a, b, c);                     // accumulate through C
    }
    int base = (wave * 16 + half * 8) * 16 + m;
#pragma unroll
    for (int r = 0; r < 8; ++r)
        hw[base + r * 16] = c[r];
}

// scalar fallback for remainder nodes (launched only when NN % 16 != 0)
__global__ void k_dense_rem(const float* __restrict__ in, const float* __restrict__ W,
                            float* __restrict__ out_, int Fin, int start, int NN) {
    int t = blockIdx.x * blockDim.x + threadIdx.x;
    int i = start + (t >> 4);
    if (i >= NN) return;
    int f = t & 15;
    float s = 0.0f;
    for (int k = 0; k < Fin; ++k) s += in[i * Fin + k] * W[k * 16 + f];
    out_[i * 16 + f] = s;
}

// ---- scatter / pointwise --------------------------------------------------

// acc[i,f] = dinv[i]^2 * hw[i,f]   (self-loop term: norm_ii = 1/deg)
__global__ void k_accinit(const float* __restrict__ hw, const float* __restrict__ dinv,
                          float* __restrict__ acc, int NN) {
    int t = blockIdx.x * blockDim.x + threadIdx.x;
    if (t >= NN * 16) return;
    float di = dinv[t >> 4];
    acc[t] = di * di * hw[t];
}

// 16 lanes per edge: coalesced gather of hw[row,*], coalesced atomics to acc[col,*]
__global__ void k_scatter(const int* __restrict__ row, const int* __restrict__ col,
                          const float* __restrict__ nrm, const float* __restrict__ hw,
                          float* __restrict__ acc, int NE) {
    int t = blockIdx.x * blockDim.x + threadIdx.x;
    if (t >= NE * 16) return;
    int e = t >> 4;
    int f = t & 15;
    float v = nrm[e] * hw[row[e] * 16 + f];
    atomicAdd(&acc[col[e] * 16 + f], v);
}

__global__ void k_biasrelu(float* __restrict__ acc, const float* __restrict__ bias,
                           int NN) {
    int t = blockIdx.x * blockDim.x + threadIdx.x;
    if (t >= NN * 16) return;
    acc[t] = fmaxf(acc[t] + bias[t & 15], 0.0f);
}

// layer-2 bias+relu fused with batch pooling (atomics into 64KB, L2-resident)
__global__ void k_biasrelupool(const float* __restrict__ acc, const float* __restrict__ bias,
                               const int* __restrict__ batch, float* __restrict__ pooled,
                               int NN) {
    int t = blockIdx.x * blockDim.x + threadIdx.x;
    if (t >= NN * 16) return;
    int i = t >> 4, f = t & 15;
    float v = fmaxf(acc[t] + bias[f], 0.0f);
    atomicAdd(&pooled[batch[i] * 16 + f], v);
}

__global__ void k_final(const float* __restrict__ pooled, const float* __restrict__ Wlin,
                        const float* __restrict__ blin, float* __restrict__ out, int G) {
    int t = blockIdx.x * blockDim.x + threadIdx.x;
    if (t >= G * 7) return;
    int g = t / 7, j = t % 7;
    float s = blin[j];
#pragma unroll
    for (int k = 0; k < 16; ++k) s += pooled[g * 16 + k] * Wlin[k * 7 + j];
    out[t] = s;
}

// ---- driver ---------------------------------------------------------------

extern "C" void kernel_launch(void* const* d_in, const int* in_sizes, int n_in,
                              void* d_out, int out_size, void* d_ws, size_t ws_size,
                              hipStream_t stream) {
    (void)n_in; (void)ws_size;
    const float* x     = (const float*)d_in[0];
    const int*   ei    = (const int*)d_in[1];
    const float* ew    = (const float*)d_in[2];
    const int*   batch = (const int*)d_in[3];
    const float* W1    = (const float*)d_in[4];
    const float* b1    = (const float*)d_in[5];
    const float* W2    = (const float*)d_in[6];
    const float* b2    = (const float*)d_in[7];
    const float* Wlin  = (const float*)d_in[8];
    const float* blin  = (const float*)d_in[9];
    float*       out   = (float*)d_out;

    const int NE = in_sizes[2];         // 3,200,000
    const int NN = in_sizes[3];         // 100,000
    const int G  = out_size / 7;        // 1000
    const int* row = ei;                // edge_index[0]
    const int* col = ei + NE;           // edge_index[1]

    // workspace layout (floats)
    float* ws     = (float*)d_ws;
    float* deg    = ws;                                  // NN (becomes dinv)
    float* nrm    = deg + NN;                            // NE
    float* B0     = nrm + NE;                            // NN*16  (hw: x@W1 / h1@W2)
    float* B1     = B0 + (size_t)NN * 16;                // NN*16  (acc1 -> h1)
    float* B2     = B1 + (size_t)NN * 16;                // NN*16  (acc2 -> h2)
    float* pooled = B2 + (size_t)NN * 16;                // G*16

    const int T = 256;
    auto cdiv = [](long long a, long long b) { return (int)((a + b - 1) / b); };
    const int ntiles = NN / 16;                          // complete tiles (6250, exact)
    const int remN   = NN - ntiles * 16;                 // 0 for this workload
    const int rstart = ntiles * 16;

    // norm precompute
    k_init <<<cdiv(NN, T), T, 0, stream>>>(deg, pooled, NN, G * 16);
    k_deg  <<<cdiv(NE, T), T, 0, stream>>>(col, ew, deg, NE);
    k_dinv <<<cdiv(NN, T), T, 0, stream>>>(deg, NN);
    k_norm <<<cdiv(NE, T), T, 0, stream>>>(row, col, ew, deg, nrm, NE);

    // layer 1
    k_dense1  <<<cdiv((long long)ntiles * 32, T), T, 0, stream>>>(x, W1, B0, ntiles);
    if (remN)
        k_dense_rem<<<cdiv((long long)remN * 16, T), T, 0, stream>>>(x, W1, B0, 3, rstart, NN);
    k_accinit <<<cdiv((long long)NN * 16, T), T, 0, stream>>>(B0, deg, B1, NN);
    k_scatter <<<cdiv((long long)NE * 16, T), T, 0, stream>>>(row, col, nrm, B0, B1, NE);
    k_biasrelu<<<cdiv((long long)NN * 16, T), T, 0, stream>>>(B1, b1, NN);

    // layer 2
    k_dense2  <<<cdiv((long long)ntiles * 32, T), T, 0, stream>>>(B1, W2, B0, ntiles);
    if (remN)
        k_dense_rem<<<cdiv((long long)remN * 16, T), T, 0, stream>>>(B1, W2, B0, 16, rstart, NN);
    k_accinit <<<cdiv((long long)NN * 16, T), T, 0, stream>>>(B0, deg, B2, NN);
    k_scatter <<<cdiv((long long)NE * 16, T), T, 0, stream>>>(row, col, nrm, B0, B2, NE);
    k_biasrelupool<<<cdiv((long long)NN * 16, T), T, 0, stream>>>(B2, b2, batch, pooled, NN);

    // head
    k_final <<<cdiv((long long)G * 7, T), T, 0, stream>>>(pooled, Wlin, blin, out, G);
}